// EnhancedGNN_76828374991280
// MI455X (gfx1250) — compile-verified
//
#include <hip/hip_runtime.h>
#include <cstdint>

// ---- static problem config (matches reference) ----
#define N_NODES   100000
#define N_EDGES   1600000
#define N_GRAPHS  100
#define NPG       1000
#define EPG       (N_EDGES / N_GRAPHS)   // 16000 edges per graph, contiguous
#define GRID_W    32
#define IN_DIM    64
#define POS_DIM   16
#define IN0       (IN_DIM + POS_DIM)     // 80
#define HID       16
#define HEADS     4
#define HD        64
#define OUT_DIM   10
#define NEG_SLOPE 0.2f
#define BN_EPS    1e-5f

typedef float v2f __attribute__((ext_vector_type(2)));
typedef float v8f __attribute__((ext_vector_type(8)));

// =====================================================================
// WMMA dense matmul: out[N,64] = in[N,K] @ W[K,64] (+ bias)
// One wave per 16-row tile; 4 column blocks of 16 -> 4x v8f accumulators.
// Uses V_WMMA_F32_16X16X4_F32 (fp32 A/B, fp32 accum).
// =====================================================================
__global__ __launch_bounds__(256) void k_linear(const float* __restrict__ in,
                                                const float* __restrict__ W,
                                                const float* __restrict__ bias,
                                                float* __restrict__ out,
                                                int N, int K) {
  int gt   = blockIdx.x * blockDim.x + threadIdx.x;
  int wave = gt >> 5;
  int lane = gt & 31;
  int tiles = N >> 4;
  if (wave >= tiles) return;

  int col0  = lane & 15;          // N index within 16-col block / M row index for A
  int khalf = (lane >> 4) << 1;   // lanes 16-31 hold K+2,K+3

  v8f acc[4];
#pragma unroll
  for (int c = 0; c < 4; ++c) {
    float bv = bias ? bias[c * 16 + col0] : 0.f;
    acc[c] = (v8f){bv, bv, bv, bv, bv, bv, bv, bv};
  }

  const float* arow = in + (size_t)((wave << 4) + col0) * K;
  for (int k0 = 0; k0 < K; k0 += 4) {
    v2f a;
    a.x = arow[k0 + khalf];
    a.y = arow[k0 + khalf + 1];
#pragma unroll
    for (int c = 0; c < 4; ++c) {
      int col = c * 16 + col0;
      v2f b;
      b.x = W[(k0 + khalf) * HD + col];
      b.y = W[(k0 + khalf + 1) * HD + col];
      acc[c] = __builtin_amdgcn_wmma_f32_16x16x4_f32(
          false, a, false, b, (short)0, acc[c], false, false);
    }
  }

  int mhi = (lane >> 4) << 3;     // lanes 16-31 hold rows M+8
#pragma unroll
  for (int c = 0; c < 4; ++c) {
#pragma unroll
    for (int i = 0; i < 8; ++i) {
      int m = (wave << 4) + i + mhi;
      out[(size_t)m * HD + c * 16 + col0] = acc[c][i];
    }
  }
}

// =====================================================================
// h0[N,80] = [ x | pos @ pe_W + pe_b ]
// =====================================================================
__global__ void k_embed(const float* __restrict__ x,
                        const float* __restrict__ peW,
                        const float* __restrict__ peb,
                        float* __restrict__ h0) {
  int t = blockIdx.x * blockDim.x + threadIdx.x;   // N*20 tasks, 4 floats each
  if (t >= N_NODES * 20) return;
  int n = t / 20, q = t % 20;
  float* o = h0 + (size_t)n * IN0;
  if (q < 16) {
    ((float4*)o)[q] = ((const float4*)(x + (size_t)n * IN_DIM))[q];
  } else {
    int ln = n % NPG;
    float px = (float)(ln / GRID_W) * (1.f / (GRID_W - 1));
    float py = (float)(ln % GRID_W) * (1.f / (GRID_W - 1));
    int j0 = (q - 16) * 4;
#pragma unroll
    for (int j = 0; j < 4; ++j) {
      int jj = j0 + j;
      o[IN_DIM + jj] = px * peW[jj] + py * peW[POS_DIM + jj] + peb[jj];
    }
  }
}

// =====================================================================
// Per-node attention dots: sAll[n,16] = [sp(4) dp(4) sd(4) dd(4)]
// =====================================================================
__global__ void k_attn(const float* __restrict__ hp, const float* __restrict__ hd,
                       const float* __restrict__ pas, const float* __restrict__ pad_,
                       const float* __restrict__ das, const float* __restrict__ dad_,
                       float* __restrict__ sAll) {
  int t = blockIdx.x * blockDim.x + threadIdx.x;
  if (t >= N_NODES * 16) return;
  int n = t >> 4, k = t & 15;
  int which = k >> 2, h = k & 3;
  const float* base = ((which < 2) ? hp : hd) + (size_t)n * HD + h * HID;
  const float* av;
  switch (which) { case 0: av = pas; break; case 1: av = pad_; break;
                   case 2: av = das; break; default: av = dad_; }
  av += h * HID;
  float s = 0.f;
#pragma unroll
  for (int j = 0; j < HID; ++j) s += base[j] * av[j];
  sAll[(size_t)n * 16 + k] = s;
}

// ---- float <-> order-preserving u32 (for LDS atomic max on floats) ----
__device__ __forceinline__ unsigned ordf(float f) {
  unsigned b = __float_as_uint(f);
  return (b & 0x80000000u) ? ~b : (b | 0x80000000u);
}
__device__ __forceinline__ float unordf(unsigned o) {
  return (o & 0x80000000u) ? __uint_as_float(o & 0x7fffffffu) : __uint_as_float(~o);
}

__device__ __forceinline__ float edge_e(int s, int d, int h,
                                        const float* __restrict__ sAll, int sOff,
                                        const float* __restrict__ posW,
                                        const float* __restrict__ posb,
                                        int usePos, int g) {
  float e = sAll[(size_t)s * 16 + sOff + h] + sAll[(size_t)d * 16 + sOff + 4 + h];
  if (usePos) {
    int ls = s - g * NPG, ld = d - g * NPG;
    float rx = (float)(ld / GRID_W - ls / GRID_W) * (1.f / (GRID_W - 1));
    float ry = (float)(ld % GRID_W - ls % GRID_W) * (1.f / (GRID_W - 1));
    e += rx * posW[h] + ry * posW[HEADS + h] + posb[h];
  }
  return e > 0.f ? e : NEG_SLOPE * e;   // leaky relu
}

// =====================================================================
// One workgroup per (graph, conv): full segment-softmax GAT aggregation
// in LDS. acc[1000*64] f32 (256KB) + max[1000*4] u32 (16KB) +
// denom[1000*4] f32 (16KB) = 288KB dynamic LDS (CDNA5 WGP has 320KB).
// =====================================================================
__global__ __launch_bounds__(1024, 1) void k_edge_conv(
    const int* __restrict__ srcI, const int* __restrict__ dstI,
    const float* __restrict__ hT, const float* __restrict__ sAll, int sOff,
    const float* __restrict__ posW, const float* __restrict__ posb, int usePos,
    float* __restrict__ out, int outStride) {
  extern __shared__ float smem[];
  float*    acc  = smem;                                  // NPG*HD
  unsigned* umax = (unsigned*)(smem + NPG * HD);          // NPG*HEADS
  float*    den  = (float*)(umax + NPG * HEADS);          // NPG*HEADS

  int g   = blockIdx.x;
  int tid = threadIdx.x;
  int ebase = g * EPG;

  for (int i = tid; i < NPG * HD; i += blockDim.x) acc[i] = 0.f;
  for (int i = tid; i < NPG * HEADS; i += blockDim.x) { umax[i] = 0u; den[i] = 0.f; }
  __syncthreads();

  // pass 1: segment max per (dst, head)
  for (int t = tid; t < EPG * HEADS; t += blockDim.x) {
    int el = t >> 2, h = t & 3;
    int s = srcI[ebase + el], d = dstI[ebase + el];
    float e = edge_e(s, d, h, sAll, sOff, posW, posb, usePos, g);
    atomicMax(&umax[(d - g * NPG) * HEADS + h], ordf(e));
  }
  __syncthreads();

  // pass 2: exp-sum denominator
  for (int t = tid; t < EPG * HEADS; t += blockDim.x) {
    int el = t >> 2, h = t & 3;
    int s = srcI[ebase + el], d = dstI[ebase + el];
    int ld = d - g * NPG;
    float e = edge_e(s, d, h, sAll, sOff, posW, posb, usePos, g);
    float p = __expf(e - unordf(umax[ld * HEADS + h]));
    atomicAdd(&den[ld * HEADS + h], p);
  }
  __syncthreads();

  // pass 3: weighted scatter of source features (LDS float atomics)
  for (int t = tid; t < EPG * HEADS; t += blockDim.x) {
    int el = t >> 2, h = t & 3;
    int s = srcI[ebase + el], d = dstI[ebase + el];
    int ld = d - g * NPG;
    float e = edge_e(s, d, h, sAll, sOff, posW, posb, usePos, g);
    float p = __expf(e - unordf(umax[ld * HEADS + h]));
    float alpha = p / (den[ld * HEADS + h] + 1e-16f);
    const float4* hv = (const float4*)(hT + (size_t)s * HD + h * HID);
    float* ab = acc + ld * HD + h * HID;
#pragma unroll
    for (int q = 0; q < 4; ++q) {
      float4 v = hv[q];
      atomicAdd(&ab[q * 4 + 0], alpha * v.x);
      atomicAdd(&ab[q * 4 + 1], alpha * v.y);
      atomicAdd(&ab[q * 4 + 2], alpha * v.z);
      atomicAdd(&ab[q * 4 + 3], alpha * v.w);
    }
  }
  __syncthreads();

  // writeout (coalesced)
  for (int i = tid; i < NPG * HD; i += blockDim.x) {
    int ln = i / HD, ch = i % HD;
    out[(size_t)(g * NPG + ln) * outStride + ch] = acc[i];
  }
}

// =====================================================================
// BN column statistics: stats[0..63]=sum, stats[64..127]=sumsq
// =====================================================================
__global__ void k_zero(float* __restrict__ p, int n) {
  int t = blockIdx.x * blockDim.x + threadIdx.x;
  if (t < n) p[t] = 0.f;
}

__global__ __launch_bounds__(256) void k_colstats(const float* __restrict__ fused,
                                                  float* __restrict__ stats) {
  __shared__ float ss[256], sq[256];
  int ch = threadIdx.x & 63, grp = threadIdx.x >> 6;
  float s = 0.f, q = 0.f;
  for (int r = blockIdx.x * 4 + grp; r < N_NODES; r += gridDim.x * 4) {
    float v = fused[(size_t)r * HD + ch];
    s += v; q += v * v;
  }
  ss[threadIdx.x] = s; sq[threadIdx.x] = q;
  __syncthreads();
  if (grp == 0) {
    s = ss[ch] + ss[ch + 64] + ss[ch + 128] + ss[ch + 192];
    q = sq[ch] + sq[ch + 64] + sq[ch + 128] + sq[ch + 192];
    atomicAdd(&stats[ch], s);
    atomicAdd(&stats[64 + ch], q);
  }
}

__global__ void k_bn_elu(const float* __restrict__ fused, const float* __restrict__ stats,
                         const float* __restrict__ gg, const float* __restrict__ bb,
                         const float* __restrict__ res, float* __restrict__ out) {
  int t = blockIdx.x * blockDim.x + threadIdx.x;
  if (t >= N_NODES * HD) return;
  int ch = t & 63;
  float mu  = stats[ch] * (1.f / N_NODES);
  float var = stats[64 + ch] * (1.f / N_NODES) - mu * mu;
  float v = (fused[t] - mu) * rsqrtf(var + BN_EPS) * gg[ch] + bb[ch];
  float e = v > 0.f ? v : (__expf(v) - 1.f);   // ELU
  out[t] = e + (res ? res[t] : 0.f);
}

// =====================================================================
// Mean pool per graph + task head
// =====================================================================
__global__ __launch_bounds__(256) void k_pool(const float* __restrict__ h2,
                                              float* __restrict__ pooled) {
  __shared__ float ss[256];
  int g = blockIdx.x;
  int ch = threadIdx.x & 63, grp = threadIdx.x >> 6;
  float s = 0.f;
  for (int n = grp; n < NPG; n += 4) s += h2[(size_t)(g * NPG + n) * HD + ch];
  ss[threadIdx.x] = s;
  __syncthreads();
  if (grp == 0)
    pooled[g * HD + ch] = (ss[ch] + ss[ch + 64] + ss[ch + 128] + ss[ch + 192]) * (1.f / NPG);
}

__global__ void k_task(const float* __restrict__ pooled, const float* __restrict__ tW,
                       const float* __restrict__ tb, float* __restrict__ out) {
  int t = blockIdx.x * blockDim.x + threadIdx.x;
  if (t >= N_GRAPHS * OUT_DIM) return;
  int g = t / OUT_DIM, o = t % OUT_DIM;
  float s = tb[o];
#pragma unroll 8
  for (int k = 0; k < HD; ++k) s += pooled[g * HD + k] * tW[k * OUT_DIM + o];
  out[t] = s;
}

// =====================================================================
// Head-diversity loss for both dW matrices -> d_out[1000]
// =====================================================================
__global__ __launch_bounds__(256) void k_div(const float* __restrict__ W0, int in0,
                                             const float* __restrict__ W1, int in1,
                                             float* __restrict__ out) {
  __shared__ float red[256];
  __shared__ float dots[10];
  __shared__ float lsum;
  if (threadIdx.x == 0) lsum = 0.f;
  const float* Ws[2] = {W0, W1};
  int ins[2] = {in0, in1};
  const int pa[10] = {0, 1, 2, 3, 0, 0, 0, 1, 1, 2};
  const int pb[10] = {0, 1, 2, 3, 1, 2, 3, 2, 3, 3};

  for (int w = 0; w < 2; ++w) {
    const float* W = Ws[w];
    int len = ins[w] * HID;
    float part[10];
#pragma unroll
    for (int d = 0; d < 10; ++d) part[d] = 0.f;
    for (int e = threadIdx.x; e < len; e += 256) {
      int i = e / HID, j = e % HID;
      float w0 = W[i * HD + 0 * HID + j];
      float w1 = W[i * HD + 1 * HID + j];
      float w2 = W[i * HD + 2 * HID + j];
      float w3 = W[i * HD + 3 * HID + j];
      part[0] += w0 * w0; part[1] += w1 * w1; part[2] += w2 * w2; part[3] += w3 * w3;
      part[4] += w0 * w1; part[5] += w0 * w2; part[6] += w0 * w3;
      part[7] += w1 * w2; part[8] += w1 * w3; part[9] += w2 * w3;
    }
#pragma unroll
    for (int d = 0; d < 10; ++d) {
      __syncthreads();
      red[threadIdx.x] = part[d];
      __syncthreads();
      for (int off = 128; off > 0; off >>= 1) {
        if (threadIdx.x < off) red[threadIdx.x] += red[threadIdx.x + off];
        __syncthreads();
      }
      if (threadIdx.x == 0) dots[d] = red[0];
    }
    __syncthreads();
    if (threadIdx.x == 0) {
      float nrm[4];
      for (int h = 0; h < 4; ++h) nrm[h] = sqrtf(dots[h]) + 1e-12f;
      float off2 = 0.f;
      for (int h = 0; h < 4; ++h) {                  // diagonal of (G - I)
        float gd = dots[h] / (nrm[h] * nrm[h]) - 1.f;
        off2 += gd * gd;
      }
      for (int d = 4; d < 10; ++d) {                 // symmetric off-diagonals
        float gv = dots[d] / (nrm[pa[d]] * nrm[pb[d]]);
        off2 += 2.f * gv * gv;
      }
      lsum += 0.1f * off2 / (float)(HEADS * (HEADS - 1));
    }
    __syncthreads();
  }
  if (threadIdx.x == 0) out[N_GRAPHS * OUT_DIM] = lsum;
}

// =====================================================================
// Host-side orchestration
// =====================================================================
extern "C" void kernel_launch(void* const* d_in, const int* in_sizes, int n_in,
                              void* d_out, int out_size, void* d_ws, size_t ws_size,
                              hipStream_t stream) {
  (void)in_sizes; (void)n_in; (void)out_size; (void)ws_size;

  const float* x    = (const float*)d_in[0];
  const int*   eidx = (const int*)d_in[1];
  // d_in[2] = batch (implicit: node/1000)
  const float* peW  = (const float*)d_in[3];
  const float* peb  = (const float*)d_in[4];
  const float* L[2][12];
  for (int l = 0; l < 2; ++l)
    for (int k = 0; k < 12; ++k) L[l][k] = (const float*)d_in[5 + l * 12 + k];
  const float* resW = (const float*)d_in[29];
  const float* resb = (const float*)d_in[30];
  const float* tW   = (const float*)d_in[31];
  const float* tb   = (const float*)d_in[32];
  const int* srcI = eidx;
  const int* dstI = eidx + N_EDGES;
  float* out = (float*)d_out;

  // workspace carve-up (floats)
  float* ws     = (float*)d_ws;
  const size_t N = N_NODES;
  float* h0     = ws;                      // N*80
  float* h1     = h0 + N * IN0;            // N*64  (layer0 output, kept for residual)
  float* hp     = h1 + N * HD;             // N*64  (pos-transformed; reused as fused)
  float* hd     = hp + N * HD;             // N*64  (div-transformed; reused as residual)
  float* sA     = hd + N * HD;             // N*16  (attention dots)
  float* cat    = sA + N * 16;             // N*128 (concat pos|div; reused as h2)
  float* pooled = cat + N * 128;           // 100*64
  float* stats  = pooled + N_GRAPHS * HD;  // 128

  const int lds_bytes = NPG * HD * 4 + NPG * HEADS * 4 + NPG * HEADS * 4;  // 288000
  (void)hipFuncSetAttribute((const void*)k_edge_conv,
                            hipFuncAttributeMaxDynamicSharedMemorySize, lds_bytes);

  k_embed<<<(N_NODES * 20 + 255) / 256, 256, 0, stream>>>(x, peW, peb, h0);

  const int tiles = N_NODES / 16;                // 6250
  const int mmblk = (tiles + 7) / 8;             // 8 waves per 256-thread block

  const float* hin = h0;
  int K = IN0;
  for (int l = 0; l < 2; ++l) {
    const float *pW = L[l][0], *pas = L[l][1], *pad_ = L[l][2],
                *posW = L[l][3], *posb = L[l][4],
                *dW = L[l][5], *das = L[l][6], *dad_ = L[l][7],
                *fW = L[l][8], *fb = L[l][9], *gg = L[l][10], *bb = L[l][11];

    k_linear<<<mmblk, 256, 0, stream>>>(hin, pW, nullptr, hp, N_NODES, K);
    k_linear<<<mmblk, 256, 0, stream>>>(hin, dW, nullptr, hd, N_NODES, K);
    k_attn<<<(N_NODES * 16 + 255) / 256, 256, 0, stream>>>(hp, hd, pas, pad_, das, dad_, sA);

    // GAT aggregation, one workgroup per graph per conv (288KB LDS each)
    k_edge_conv<<<N_GRAPHS, 1024, lds_bytes, stream>>>(srcI, dstI, hp, sA, 0,
                                                       posW, posb, 1, cat, 2 * HD);
    k_edge_conv<<<N_GRAPHS, 1024, lds_bytes, stream>>>(srcI, dstI, hd, sA, 8,
                                                       nullptr, nullptr, 0, cat + HD, 2 * HD);

    // fused = cat @ fW + fb  (K=128) -> reuse hp
    k_linear<<<mmblk, 256, 0, stream>>>(cat, fW, fb, hp, N_NODES, 2 * HD);

    k_zero<<<1, 128, 0, stream>>>(stats, 128);
    k_colstats<<<256, 256, 0, stream>>>(hp, stats);

    const float* res = nullptr;
    float* hout = (l == 0) ? h1 : cat;           // h2 reuses cat region
    if (l == 1) {
      k_linear<<<mmblk, 256, 0, stream>>>(h1, resW, resb, hd, N_NODES, HD);  // residual -> hd
      res = hd;
    }
    k_bn_elu<<<(N_NODES * HD + 255) / 256, 256, 0, stream>>>(hp, stats, gg, bb, res, hout);

    hin = hout;
    K = HD;
  }

  k_pool<<<N_GRAPHS, 256, 0, stream>>>(cat, pooled);
  k_task<<<(N_GRAPHS * OUT_DIM + 255) / 256, 256, 0, stream>>>(pooled, tW, tb, out);
  k_div<<<1, 256, 0, stream>>>(L[0][5], IN0, L[1][5], HD, out);
}